// HashEncoding_22771916603453
// MI455X (gfx1250) — compile-verified
//
#include <hip/hip_runtime.h>
#include <cmath>

typedef unsigned int v4u __attribute__((ext_vector_type(4)));
typedef int          v8i __attribute__((ext_vector_type(8)));
typedef int          v4i __attribute__((ext_vector_type(4)));
typedef float        v2f __attribute__((ext_vector_type(2)));

#define N_LEVELS 16

struct LevelParams {
    float    scale[N_LEVELS];
    unsigned size[N_LEVELS];
    unsigned res[N_LEVELS];   // 0 => hashed level
};
struct TablePtrs { const v2f* t[N_LEVELS]; };

// One level per blockIdx.y -> all level params are wave-uniform (SGPRs),
// hash-vs-linear branch is uniform, and LDS-vs-global table source is a
// uniform branch (kept as separate loops so the LDS path stays addrspace(3)
// and lowers to ds_load_b64 instead of flat loads).
__global__ __launch_bounds__(256) void hashenc_kernel(
    const float* __restrict__ in, TablePtrs tabs, LevelParams P,
    float* __restrict__ out, unsigned npts)
{
    __shared__ v2f tab0[4096];                       // level-0 table: 4096 x float2 = 32 KB
    const unsigned L = blockIdx.y;

    if (L == 0) {
        // Stage the full level-0 table into LDS with the Tensor Data Mover.
        // D# per CDNA5 ISA §8: 1D tile of 8192 f32 (data_size=4B code 2).
        if (threadIdx.x < 32) {                      // first wave only (uniform branch)
            unsigned lds = (unsigned)(size_t)(&tab0[0]);          // addr[31:0] = LDS offset
            unsigned long long ga = (unsigned long long)(size_t)tabs.t[0];
            v4u g0 = { 1u,                                        // count=1, user descriptor
                       lds,                                       // lds_addr
                       (unsigned)(ga & 0xFFFFFFFFu),              // global_addr[31:0]
                       ((unsigned)(ga >> 32) & 0x1FFFFFFu) | (2u << 30) }; // addr[56:32] | type=2
            v8i g1 = { (int)(2u << 16),          // data_size=4B; wg_mask=0 (not in cluster)
                       (int)(8192u << 16),       // tensor_dim0[15:0]=8192 in bits[63:48]
                       (int)(1u << 16),          // tensor_dim0 hi=0; tensor_dim1=1
                       (int)(8192u << 16),       // tile_dim0 = 8192
                       1,                        // tile_dim1 = 1; tile_dim2 = 0
                       8192,                     // tensor_dim0_stride[31:0]
                       (int)(8192u << 16),       // stride0 hi=0; tensor_dim1_stride[15:0]
                       0 };                      // tensor_dim1_stride[47:16]
            v4i g2 = {0,0,0,0}, g3 = {0,0,0,0};
            v8i g4 = {0,0,0,0,0,0,0,0};          // unused trailing group (6-arg builtin)
            __builtin_amdgcn_tensor_load_to_lds(g0, g1, g2, g3, g4, 0);
            __builtin_amdgcn_s_wait_tensorcnt(0);
        }
        __syncthreads();
    }

    unsigned p = blockIdx.x * 256u + threadIdx.x;
    if (p >= npts) return;

    float ix = in[3u * p + 0u], iy = in[3u * p + 1u], iz = in[3u * p + 2u];

    float    s    = P.scale[L];
    unsigned size = P.size[L];
    unsigned res  = P.res[L];
    bool     hashed = (res == 0u);
    unsigned mask = size - 1u;     // hashed levels: size = 2^19
    unsigned res2 = res * res;

    float px = ix * s + 0.5f, py = iy * s + 0.5f, pz = iz * s + 0.5f;
    float fx = floorf(px), fy = floorf(py), fz = floorf(pz);
    float dx = px - fx, dy = py - fy, dz = pz - fz;
    unsigned gx = (unsigned)(int)fx, gy = (unsigned)(int)fy, gz = (unsigned)(int)fz;
    float wx[2] = {1.0f - dx, dx};
    float wy[2] = {1.0f - dy, dy};
    float wz[2] = {1.0f - dz, dz};

    // Compute all 8 corner indices + trilinear weights first (shared between
    // the LDS and global gather paths).
    unsigned idx[8];
    float    w[8];
#pragma unroll
    for (int cx = 0; cx < 2; ++cx) {
#pragma unroll
        for (int cy = 0; cy < 2; ++cy) {
#pragma unroll
            for (int cz = 0; cz < 2; ++cz) {
                int c = cx * 4 + cy * 2 + cz;        // CORNERS order: z fastest
                unsigned x = gx + (unsigned)cx;
                unsigned y = gy + (unsigned)cy;
                unsigned z = gz + (unsigned)cz;
                if (hashed) {
                    idx[c] = (x ^ (y * 2654435761u) ^ (z * 805459861u)) & mask;
                } else {
                    unsigned lin = x + y * res + z * res2;   // lin < 2*size always
                    idx[c] = lin - (lin >= size ? size : 0u); // exact '% size'
                }
                w[c] = wx[cx] * wy[cy] * wz[cz];
            }
        }
    }

    float a0 = 0.0f, a1 = 0.0f;
    if (L == 0) {
        // Level-0 gathers served from LDS (ds_load_b64), table staged by TDM.
#pragma unroll
        for (int c = 0; c < 8; ++c) {
            v2f f = tab0[idx[c]];
            a0 += w[c] * f.x;
            a1 += w[c] * f.y;
        }
    } else {
        const v2f* __restrict__ tab = tabs.t[L];
#pragma unroll
        for (int c = 0; c < 8; ++c) {
            v2f f = tab[idx[c]];                     // global_load_b64, L2-resident
            a0 += w[c] * f.x;
            a1 += w[c] * f.y;
        }
    }

    // Streaming 128 MB output: non-temporal so tables stay resident in L2.
    v2f r = {a0, a1};
    __builtin_nontemporal_store(r, (v2f*)(out + (size_t)p * 32u + 2u * (size_t)L));
}

extern "C" void kernel_launch(void* const* d_in, const int* in_sizes, int n_in,
                              void* d_out, int out_size, void* d_ws, size_t ws_size,
                              hipStream_t stream) {
    (void)n_in; (void)out_size; (void)d_ws; (void)ws_size;

    LevelParams P;
    TablePtrs   T;
    const double pls = 1.38191288;
    for (int i = 0; i < N_LEVELS; ++i) {
        // Mirror the Python: scale = 2^(i*log2(pls))*16 - 1 (double), res = ceil+1,
        // size = min(ceil(res^3/8)*8, 2^19), hashed if capped.
        double scale = pow(2.0, (double)i * log2(pls)) * 16.0 - 1.0;
        long long res = (long long)ceil(scale) + 1;
        unsigned long long r3 = (unsigned long long)(res * res * res);
        unsigned long long sz = ((r3 + 7ull) / 8ull) * 8ull;
        if (sz >= (1ull << 19)) { sz = 1ull << 19; P.res[i] = 0u; }
        else                    { P.res[i] = (unsigned)res; }
        P.size[i]  = (unsigned)sz;
        P.scale[i] = (float)scale;
        T.t[i] = (const v2f*)d_in[1 + i];
    }

    unsigned npts = (unsigned)(in_sizes[0] / 3);
    dim3 grid((npts + 255u) / 256u, N_LEVELS, 1);
    hashenc_kernel<<<grid, 256, 0, stream>>>((const float*)d_in[0], T, P,
                                             (float*)d_out, npts);
}